// CustomCrossEntropyLoss_38732015075867
// MI455X (gfx1250) — compile-verified
//
#include <hip/hip_runtime.h>
#include <math.h>

// ---------------- problem constants ----------------
#define B_ROWS   4096
#define VCOLS    50257
#define BDIM     256
#define WAVES    (BDIM / 32)
#define TILE_F4  1024                 // float4s per tile = BDIM * 4
#define TILE_F   (TILE_F4 * 4)        // 4096 floats = 16 KB
#define NTILES   12                   // 12 * 4096 = 49152 floats via async path

typedef __attribute__((ext_vector_type(2))) float v2f;
typedef __attribute__((ext_vector_type(8))) float v8f;

// ---- per-wave s_wait_asynccnt (immediate form) ----
#define S_WAIT_ASYNC(n) asm volatile("s_wait_asynccnt " #n ::: "memory")

// ---- async global->LDS copy, 16B per lane (GVS addressing: saddr + 32b voffset) ----
__device__ __forceinline__ void async_ld_f4(unsigned int lds_off,
                                            const float* sbase,
                                            unsigned int gbyte_off) {
  asm volatile("global_load_async_to_lds_b128 %0, %1, %2"
               :: "v"(lds_off), "v"(gbyte_off), "s"(sbase)
               : "memory");
}

// ---- 32-lane fp32 sum via V_WMMA_F32_16X16X4_F32 (B = ones) ----
// A.v0 = x(lane), A.v1 = 0  =>  D[m][n] = x[m] + x[m+16]
// lane sums its 8 D registers, then adds the lane^16 partner: full wave sum.
__device__ __forceinline__ float wave_sum_wmma(float x) {
  v2f a; a.x = x;   a.y = 0.0f;
  v2f b; b.x = 1.0f; b.y = 1.0f;
  v8f c = {};
  c = __builtin_amdgcn_wmma_f32_16x16x4_f32(false, a, false, b,
                                            (short)0, c, false, false);
  float t = c[0] + c[1] + c[2] + c[3] + c[4] + c[5] + c[6] + c[7];
  t += __shfl_xor(t, 16, 32);
  return t;
}

// online logsumexp accumulator update with a pre-computed chunk max
__device__ __forceinline__ void lse_chunk4(float& m, float& s, float4 v) {
  float c  = fmaxf(fmaxf(v.x, v.y), fmaxf(v.z, v.w));
  float nm = fmaxf(m, c);
  s = s * __expf(m - nm)
    + __expf(v.x - nm) + __expf(v.y - nm) + __expf(v.z - nm) + __expf(v.w - nm);
  m = nm;
}

__device__ __forceinline__ void lse_scalar(float& m, float& s, float x) {
  float nm = fmaxf(m, x);
  s = s * __expf(m - nm) + __expf(x - nm);
  m = nm;
}

// ---------------- kernel 1: one block per row ----------------
__global__ __launch_bounds__(BDIM)
void ce_row_kernel(const float* __restrict__ labels,
                   const int*   __restrict__ truth,
                   float*       __restrict__ row_loss) {
  __shared__ __align__(16) float smem[2][TILE_F];
  __shared__ float red_m[WAVES];
  __shared__ float red_s[WAVES];

  const int tid = threadIdx.x;
  const int row = blockIdx.x;
  const float* rowp = labels + (size_t)row * VCOLS;

  // peel 0..3 elements so async b128 loads are 16B aligned
  const size_t e0 = (size_t)row * VCOLS;
  const int p = (int)((4u - ((unsigned)e0 & 3u)) & 3u);
  const float* abase = rowp + p;                 // 16B aligned
  const int nf4 = (VCOLS - p) >> 2;              // aligned float4 count (>= NTILES*TILE_F4)

  unsigned int lds0 = (unsigned int)(uintptr_t)&smem[0][0];
  unsigned int lds1 = (unsigned int)(uintptr_t)&smem[1][0];

  auto issue_tile = [&](int t) {
    unsigned int g0 = (unsigned int)(t * TILE_F4) * 16u;
    unsigned int lb = (t & 1) ? lds1 : lds0;
#pragma unroll
    for (int k = 0; k < 4; ++k) {
      unsigned int f4 = (unsigned int)(k * BDIM + tid);
      async_ld_f4(lb + f4 * 16u, abase, g0 + f4 * 16u);
    }
  };

  float m = -INFINITY, s = 0.0f;

  issue_tile(0);
  for (int t = 0; t < NTILES; ++t) {
    if (t + 1 < NTILES) { issue_tile(t + 1); S_WAIT_ASYNC(4); }
    else                {                    S_WAIT_ASYNC(0); }

    const float4* buf = (const float4*)&smem[t & 1][0];
    float4 v0 = buf[0 * BDIM + tid];
    float4 v1 = buf[1 * BDIM + tid];
    float4 v2 = buf[2 * BDIM + tid];
    float4 v3 = buf[3 * BDIM + tid];

    // 16-element chunk: one rescale exp per 16 elements
    float c = fmaxf(fmaxf(fmaxf(fmaxf(v0.x, v0.y), fmaxf(v0.z, v0.w)),
                          fmaxf(fmaxf(v1.x, v1.y), fmaxf(v1.z, v1.w))),
                    fmaxf(fmaxf(fmaxf(v2.x, v2.y), fmaxf(v2.z, v2.w)),
                          fmaxf(fmaxf(v3.x, v3.y), fmaxf(v3.z, v3.w))));
    float nm = fmaxf(m, c);
    float acc =
        __expf(v0.x - nm) + __expf(v0.y - nm) + __expf(v0.z - nm) + __expf(v0.w - nm)
      + __expf(v1.x - nm) + __expf(v1.y - nm) + __expf(v1.z - nm) + __expf(v1.w - nm)
      + __expf(v2.x - nm) + __expf(v2.y - nm) + __expf(v2.z - nm) + __expf(v2.w - nm)
      + __expf(v3.x - nm) + __expf(v3.y - nm) + __expf(v3.z - nm) + __expf(v3.w - nm);
    s = s * __expf(m - nm) + acc;
    m = nm;
  }

  // prologue scalars (misalignment peel)
  if (tid < p) lse_scalar(m, s, rowp[tid]);

  // tail float4s beyond the async-tiled region (direct, aligned)
  const float4* a4 = (const float4*)abase;
  for (int f4 = NTILES * TILE_F4 + tid; f4 < nf4; f4 += BDIM)
    lse_chunk4(m, s, a4[f4]);

  // tail scalars
  {
    const int start = p + (nf4 << 2);
    const int r = VCOLS - start;          // 0..3
    if (tid < r) lse_scalar(m, s, rowp[start + tid]);
  }

  // ---- block reduction: max first, then WMMA wave sums of rescaled s ----
  const int lane = tid & 31, w = tid >> 5;
  float wm = m;
#pragma unroll
  for (int o = 16; o > 0; o >>= 1) wm = fmaxf(wm, __shfl_xor(wm, o, 32));
  if (lane == 0) red_m[w] = wm;
  __syncthreads();
  float M = red_m[0];
#pragma unroll
  for (int i = 1; i < WAVES; ++i) M = fmaxf(M, red_m[i]);

  float sp   = s * __expf(m - M);
  float wsum = wave_sum_wmma(sp);          // EXEC all-1s here: uniform path
  if (lane == 0) red_s[w] = wsum;
  __syncthreads();

  if (tid == 0) {
    float S = 0.0f;
#pragma unroll
    for (int i = 0; i < WAVES; ++i) S += red_s[i];
    float lse    = M + logf(S);
    float picked = rowp[truth[row]];
    row_loss[row] = lse - picked;
  }
}

// ---------------- kernel 2: mean over 4096 row losses ----------------
__global__ __launch_bounds__(BDIM)
void ce_mean_kernel(const float* __restrict__ row_loss, float* __restrict__ out) {
  __shared__ float red[WAVES];
  const int tid = threadIdx.x;
  float s = 0.0f;
  for (int i = tid; i < B_ROWS; i += BDIM) s += row_loss[i];
  float wsum = wave_sum_wmma(s);
  const int lane = tid & 31, w = tid >> 5;
  if (lane == 0) red[w] = wsum;
  __syncthreads();
  if (tid == 0) {
    float tot = 0.0f;
#pragma unroll
    for (int i = 0; i < WAVES; ++i) tot += red[i];
    out[0] = tot * (1.0f / (float)B_ROWS);
  }
}

extern "C" void kernel_launch(void* const* d_in, const int* in_sizes, int n_in,
                              void* d_out, int out_size, void* d_ws, size_t ws_size,
                              hipStream_t stream) {
  const float* labels = (const float*)d_in[0];
  const int*   truth  = (const int*)d_in[1];
  float*       out    = (float*)d_out;
  float*       rloss  = (float*)d_ws;    // 4096 floats of scratch

  ce_row_kernel<<<B_ROWS, BDIM, 0, stream>>>(labels, truth, rloss);
  ce_mean_kernel<<<1, BDIM, 0, stream>>>(rloss, out);
}